// OhemCrossEntropy2d_4312147165866
// MI455X (gfx1250) — compile-verified
//
#include <hip/hip_runtime.h>
#include <hip/hip_bf16.h>
#include <stdint.h>

// ---------------- problem constants ----------------
#define IGNORE_LBL 255
#define NCLS 19
#define HW   262144u          // 512*512 = 2^18
#define PIX  2097152u         // 8 * HW
#define KMIN 200000u
#define THRESH_F 0.6f
#define NBR  2048             // reduction partial blocks

// ---------------- workspace layout (bytes) ----------------
// [0,8MB)      prob bits (uint, NaN marker 0x7FFFFFFF for invalid)
// [8MB,16MB)   nll (float)
// then: hist1[2048], hist2[2048], hist3[1024], state[256B], psum[2048], pcnt[2048]
static const size_t OFF_PROB = 0;
static const size_t OFF_NLL  = (size_t)PIX * 4;
static const size_t OFF_H1   = OFF_NLL + (size_t)PIX * 4;
static const size_t OFF_H2   = OFF_H1 + 2048 * 4;
static const size_t OFF_H3   = OFF_H2 + 2048 * 4;
static const size_t OFF_ST   = OFF_H3 + 1024 * 4;
static const size_t OFF_PSUM = OFF_ST + 256;
static const size_t OFF_PCNT = OFF_PSUM + (size_t)NBR * 4;
// state[]: 0=found1 1=b1 2=remk1 3=found2 4=b2 5=remk2 6=thrBits

// ---------------- CDNA5 TDM (Tensor Data Mover) ----------------
typedef unsigned int u32x4 __attribute__((ext_vector_type(4)));
typedef int          i32x8 __attribute__((ext_vector_type(8)));
typedef int          i32x4 __attribute__((ext_vector_type(4)));

#if defined(__has_builtin)
#  if __has_builtin(__builtin_amdgcn_tensor_load_to_lds)
#    define USE_TDM 1
#  endif
#endif
#ifndef USE_TDM
#  define USE_TDM 0
#endif

#if USE_TDM
// 2D tile load: tileX elems/row, tileY rows, row stride strideElems (data_size=4B)
__device__ __forceinline__ void tdm_load_tile(const float* gsrc, unsigned lds_off,
                                              unsigned tileX, unsigned tileY,
                                              unsigned strideElems)
{
  unsigned long long ga = (unsigned long long)(uintptr_t)gsrc;
  u32x4 g0;
  g0[0] = 1u;                                            // count=1, user mode, no gather
  g0[1] = lds_off;                                       // lds_addr (bytes)
  g0[2] = (unsigned)(ga & 0xFFFFFFFFull);                // global_addr[31:0]
  g0[3] = (unsigned)((ga >> 32) & 0x1FFFFFFull) | (2u << 30); // addr[56:32] | type=2

  i32x8 g1;
  g1[0] = (int)(2u << 16);                               // data_size=2 -> 4B
  g1[1] = (int)((tileX & 0xFFFFu) << 16);                // tensor_dim0[15:0] @ bits63:48
  g1[2] = (int)(((tileX >> 16) & 0xFFFFu) | ((tileY & 0xFFFFu) << 16)); // td0 hi | td1 lo
  g1[3] = (int)(((tileY >> 16) & 0xFFFFu) | ((tileX & 0xFFFFu) << 16)); // td1 hi | tile_dim0
  g1[4] = (int)(tileY & 0xFFFFu);                        // tile_dim1 | tile_dim2=0
  g1[5] = (int)strideElems;                              // tensor_dim0_stride[31:0]
  g1[6] = 0;                                             // stride0 hi | stride1 lo (unused, 2D)
  g1[7] = 0;
  i32x4 z4 = {0, 0, 0, 0};
#if defined(__clang_major__) && (__clang_major__ >= 23)
  i32x8 z8 = {0, 0, 0, 0, 0, 0, 0, 0};
  __builtin_amdgcn_tensor_load_to_lds(g0, g1, z4, z4, z8, 0);
#else
  __builtin_amdgcn_tensor_load_to_lds(g0, g1, z4, z4, 0);
#endif
}
#endif

// ---------------- pass 1: per-pixel softmax stats ----------------
// 1024 blocks x 128 threads (4 waves). Each wave per iteration handles a
// 64-pixel x 19-channel tile, TDM-staged into LDS, double-buffered.
__global__ __launch_bounds__(128)
void k_compute(const float* __restrict__ predict, const int* __restrict__ target,
               unsigned* __restrict__ probBits, float* __restrict__ nllOut)
{
  const int lane = (int)(threadIdx.x & 31u);
  const int wv   = (int)(threadIdx.x >> 5);
  const size_t blkbase = (size_t)blockIdx.x * 2048u;

#if USE_TDM
  __shared__ float tile[2][4][NCLS * 64];

  #define ISSUE(it, bsel) do {                                              \
      size_t tb_ = blkbase + (size_t)(it) * 256u + (size_t)wv * 64u;        \
      unsigned n_  = (unsigned)(tb_ >> 18);                                 \
      unsigned hw_ = (unsigned)(tb_ & (HW - 1u));                           \
      const float* gsrc_ = predict + (size_t)n_ * ((size_t)NCLS * HW) + hw_;\
      unsigned lds_ = (unsigned)(uintptr_t)(&tile[(bsel)][wv][0]);          \
      tdm_load_tile(gsrc_, lds_, 64u, (unsigned)NCLS, HW);                  \
    } while (0)

  ISSUE(0, 0);
  for (int it = 0; it < 8; ++it) {
    if (it + 1 < 8) { ISSUE(it + 1, (it + 1) & 1); __builtin_amdgcn_s_wait_tensorcnt(1); }
    else            { __builtin_amdgcn_s_wait_tensorcnt(0); }
    const float* buf = &tile[it & 1][wv][0];
    const size_t tb = blkbase + (size_t)it * 256u + (size_t)wv * 64u;
    if (tb + 256u < PIX) __builtin_prefetch((const void*)(target + tb + 256u), 0, 1);
    #pragma unroll
    for (int h = 0; h < 2; ++h) {
      const int i = lane + h * 32;
      const size_t p = tb + (size_t)i;
      const int lbl = target[p];
      const bool valid = (lbl != IGNORE_LBL);
      const int cls = valid ? lbl : 0;
      float sum = 0.f;
      #pragma unroll
      for (int c = 0; c < NCLS; ++c) sum += __expf(buf[c * 64 + i]);
      const float xt  = buf[cls * 64 + i];
      const float nll = __logf(sum) - xt;
      float pr = __expf(-nll);
      pr = fminf(pr, 1.0f);
      probBits[p] = valid ? __float_as_uint(pr) : 0x7FFFFFFFu;  // NaN marker
      nllOut[p]   = nll;
    }
  }
  #undef ISSUE
#else
  // fallback: direct coalesced global loads
  for (int it = 0; it < 8; ++it) {
    const size_t tb = blkbase + (size_t)it * 256u + (size_t)wv * 64u;
    #pragma unroll
    for (int h = 0; h < 2; ++h) {
      const int i = lane + h * 32;
      const size_t p = tb + (size_t)i;
      const unsigned n = (unsigned)(p >> 18);
      const unsigned hw = (unsigned)(p & (HW - 1u));
      const float* base = predict + (size_t)n * ((size_t)NCLS * HW) + hw;
      const int lbl = target[p];
      const bool valid = (lbl != IGNORE_LBL);
      const int cls = valid ? lbl : 0;
      float sum = 0.f;
      #pragma unroll
      for (int c = 0; c < NCLS; ++c) sum += __expf(base[(size_t)c * HW]);
      const float xt  = base[(size_t)cls * HW];
      const float nll = __logf(sum) - xt;
      float pr = __expf(-nll);
      pr = fminf(pr, 1.0f);
      probBits[p] = valid ? __float_as_uint(pr) : 0x7FFFFFFFu;
      nllOut[p]   = nll;
    }
  }
#endif
}

// ---------------- radix-select histograms (L2-resident 8MB stream) ----------------
// mode 0: bin = bits>>21 (valid only); mode 1: match hi11==b1, bin = bits[20:10];
// mode 2: match bits[31:10]==pref22, bin = bits[9:0]
__global__ __launch_bounds__(256)
void k_hist(const unsigned* __restrict__ probBits, const unsigned* __restrict__ state,
            unsigned* __restrict__ hist, int mode)
{
  if (mode > 0 && state[0] == 0u) return;   // uniform: selection resolved to +inf
  __shared__ unsigned h[2048];
  for (unsigned i = threadIdx.x; i < 2048u; i += blockDim.x) h[i] = 0u;
  __syncthreads();
  unsigned hi = 0u, pref = 0u;
  if (mode == 1) hi = state[1];
  if (mode == 2) pref = (state[1] << 11) | state[4];
  for (size_t p = (size_t)blockIdx.x * blockDim.x + threadIdx.x; p < PIX;
       p += (size_t)gridDim.x * blockDim.x) {
    const unsigned b = probBits[p];
    if (b > 0x3F800000u) continue;          // invalid marker / out of (0,1]
    if (mode == 0)      atomicAdd(&h[b >> 21], 1u);
    else if (mode == 1) { if ((b >> 21) == hi)   atomicAdd(&h[(b >> 10) & 0x7FFu], 1u); }
    else                { if ((b >> 10) == pref) atomicAdd(&h[b & 0x3FFu], 1u); }
  }
  __syncthreads();
  for (unsigned i = threadIdx.x; i < 2048u; i += blockDim.x)
    if (h[i]) atomicAdd(&hist[i], h[i]);
}

// single-block Hillis-Steele scan over 2048 bins; finds bin containing rank K
__global__ __launch_bounds__(1024)
void k_scan2048(const unsigned* __restrict__ hist, unsigned* __restrict__ state, int stage)
{
  if (stage == 1 && state[0] == 0u) return;
  const unsigned K = (stage == 0) ? KMIN : state[2];
  const int t = (int)threadIdx.x;
  __shared__ unsigned s[1024];
  const unsigned h0 = hist[2 * t], h1 = hist[2 * t + 1];
  const unsigned v = h0 + h1;
  s[t] = v; __syncthreads();
  for (int off = 1; off < 1024; off <<= 1) {
    unsigned x = (t >= off) ? s[t - off] : 0u;
    __syncthreads(); s[t] += x; __syncthreads();
  }
  const unsigned incl = s[t];
  const unsigned excl = incl - v;
  if (excl < K && K <= incl) {
    unsigned bin, before;
    if (excl + h0 >= K) { bin = 2u * (unsigned)t;      before = excl; }
    else                { bin = 2u * (unsigned)t + 1u; before = excl + h0; }
    const int base = (stage == 0) ? 0 : 3;
    state[base + 0] = 1u;
    state[base + 1] = bin;
    state[base + 2] = K - before;           // 1-based rank within bin
  }
}

// final 1024-bin scan -> exact kth prob bits -> threshold
__global__ __launch_bounds__(1024)
void k_scan_final(const unsigned* __restrict__ hist3, unsigned* __restrict__ state)
{
  const int t = (int)threadIdx.x;
  if (state[3] == 0u) {                     // fewer than K valid -> kth = +inf
    if (t == 0) state[6] = 0x7F800000u;
    return;
  }
  const unsigned K = state[5];
  __shared__ unsigned s[1024];
  const unsigned v = hist3[t];
  s[t] = v; __syncthreads();
  for (int off = 1; off < 1024; off <<= 1) {
    unsigned x = (t >= off) ? s[t - off] : 0u;
    __syncthreads(); s[t] += x; __syncthreads();
  }
  const unsigned incl = s[t], excl = incl - v;
  if (excl < K && K <= incl) {
    const unsigned kthBits = (((state[1] << 11) | state[4]) << 10) | (unsigned)t;
    const float kth = __uint_as_float(kthBits);
    const float thr = (kth > THRESH_F) ? kth : THRESH_F;
    state[6] = __float_as_uint(thr);
  }
}

// ---------------- final masked mean (deterministic fixed-order reduction) ----------------
__global__ __launch_bounds__(256)
void k_reduce(const unsigned* __restrict__ probBits, const float* __restrict__ nll,
              const unsigned* __restrict__ state, float* __restrict__ psum,
              unsigned* __restrict__ pcnt)
{
  const float thr = __uint_as_float(state[6]);
  float s = 0.f; unsigned c = 0u;
  for (size_t p = (size_t)blockIdx.x * blockDim.x + threadIdx.x; p < PIX;
       p += (size_t)gridDim.x * blockDim.x) {
    const float pr = __uint_as_float(probBits[p]);  // invalid -> NaN -> false
    if (pr <= thr) { s += nll[p]; c += 1u; }
  }
  __shared__ float fs[256]; __shared__ unsigned cs[256];
  const int t = (int)threadIdx.x;
  fs[t] = s; cs[t] = c; __syncthreads();
  for (int o = 128; o > 0; o >>= 1) {
    if (t < o) { fs[t] += fs[t + o]; cs[t] += cs[t + o]; }
    __syncthreads();
  }
  if (t == 0) { psum[blockIdx.x] = fs[0]; pcnt[blockIdx.x] = cs[0]; }
}

__global__ __launch_bounds__(1024)
void k_final(const float* __restrict__ psum, const unsigned* __restrict__ pcnt,
             float* __restrict__ out)
{
  __shared__ float fs[1024]; __shared__ unsigned cs[1024];
  const int t = (int)threadIdx.x;
  fs[t] = psum[t] + psum[t + 1024];
  cs[t] = pcnt[t] + pcnt[t + 1024];
  __syncthreads();
  for (int o = 512; o > 0; o >>= 1) {
    if (t < o) { fs[t] += fs[t + o]; cs[t] += cs[t + o]; }
    __syncthreads();
  }
  if (t == 0) {
    const unsigned n = cs[0] > 0u ? cs[0] : 1u;
    out[0] = fs[0] / (float)n;
  }
}

// ---------------- launch ----------------
extern "C" void kernel_launch(void* const* d_in, const int* in_sizes, int n_in,
                              void* d_out, int out_size, void* d_ws, size_t ws_size,
                              hipStream_t stream)
{
  (void)in_sizes; (void)n_in; (void)out_size; (void)ws_size;
  const float* predict = (const float*)d_in[0];
  const int*   target  = (const int*)d_in[1];
  char* ws = (char*)d_ws;
  unsigned* probBits = (unsigned*)(ws + OFF_PROB);
  float*    nll      = (float*)(ws + OFF_NLL);
  unsigned* h1 = (unsigned*)(ws + OFF_H1);
  unsigned* h2 = (unsigned*)(ws + OFF_H2);
  unsigned* h3 = (unsigned*)(ws + OFF_H3);
  unsigned* st = (unsigned*)(ws + OFF_ST);
  float*    psum = (float*)(ws + OFF_PSUM);
  unsigned* pcnt = (unsigned*)(ws + OFF_PCNT);

  hipMemsetAsync(ws + OFF_H1, 0, OFF_PSUM - OFF_H1, stream);  // hists + state
  k_compute<<<1024, 128, 0, stream>>>(predict, target, probBits, nll);
  k_hist<<<512, 256, 0, stream>>>(probBits, st, h1, 0);
  k_scan2048<<<1, 1024, 0, stream>>>(h1, st, 0);
  k_hist<<<512, 256, 0, stream>>>(probBits, st, h2, 1);
  k_scan2048<<<1, 1024, 0, stream>>>(h2, st, 1);
  k_hist<<<512, 256, 0, stream>>>(probBits, st, h3, 2);
  k_scan_final<<<1, 1024, 0, stream>>>(h3, st);
  k_reduce<<<NBR, 256, 0, stream>>>(probBits, nll, st, psum, pcnt);
  k_final<<<1, 1024, 0, stream>>>(psum, pcnt, (float*)d_out);
}